// InterpolationBlock2D_Lin_26010321944824
// MI455X (gfx1250) — compile-verified
//
#include <hip/hip_runtime.h>

// FE interpolation: u[d,n] = sum_k shape[n,k] * nodal[d, conn[cell_id[n],k]-1]
//   nodal_values   : (2, N_NODES)  f32   d_in[0]
//   shape_functions: (N_PTS, 3)    f32   d_in[1]
//   connectivity   : (N_ELEM, 3)   i32   d_in[2]
//   cell_id        : (N_PTS,)      i32   d_in[3]
//   out            : (2, N_PTS)    f32
//
// Memory-bound gather: ~24 B HBM / point streaming + L2-resident gathers
// (conn 12MB, nodal 4MB << 192MB L2). No matrix structure -> no WMMA.
// gfx1250-specific paths used: global_prefetch_b8 (streaming inputs) and
// non-temporal (TH=NT) b128 stores for the write-once output.

typedef float v4f __attribute__((ext_vector_type(4)));
typedef int   v4i __attribute__((ext_vector_type(4)));

__global__ __launch_bounds__(256) void interp2d_lin_kernel(
    const float* __restrict__ nodal,   // (2, n_nodes)
    const float* __restrict__ shape,   // (n_pts, 3)
    const int*   __restrict__ conn,    // (n_elem, 3)
    const int*   __restrict__ cid,     // (n_pts)
    float*       __restrict__ out,     // (2, n_pts)
    int n_pts, int n_nodes)
{
    const int stride = gridDim.x * blockDim.x * 4;   // points per grid-iteration
    int p = (blockIdx.x * blockDim.x + threadIdx.x) * 4;

    for (; p + 3 < n_pts; p += stride) {
        // Prefetch next iteration's streaming data (speculative: silently
        // dropped if past end of allocation). -> global_prefetch_b8
        __builtin_prefetch(cid + p + stride, 0, 1);
        __builtin_prefetch(shape + (size_t)(p + stride) * 3, 0, 1);

        // 16B-aligned wide loads: p % 4 == 0 -> cid+p is 16B aligned,
        // shape + 3p is a 48B-multiple offset -> 16B aligned.
        const v4i c = *(const v4i*)(cid + p);
        const v4f* sp = (const v4f*)(shape + (size_t)p * 3);
        const v4f sA = sp[0];
        const v4f sB = sp[1];
        const v4f sC = sp[2];

        const int   cc[4]   = { c.x, c.y, c.z, c.w };
        const float s[4][3] = { { sA.x, sA.y, sA.z },
                                { sA.w, sB.x, sB.y },
                                { sB.z, sB.w, sC.x },
                                { sC.y, sC.z, sC.w } };

        float u0[4], u1[4];
#pragma unroll
        for (int i = 0; i < 4; ++i) {
            // connectivity row: 12B at 12B stride (L2-resident, 3x b32)
            const int* cr = conn + (size_t)cc[i] * 3;
            const int i0 = cr[0] - 1;
            const int i1 = cr[1] - 1;
            const int i2 = cr[2] - 1;
            // nodal gathers (4MB table, L2-resident)
            const float a0 = nodal[i0];
            const float a1 = nodal[i1];
            const float a2 = nodal[i2];
            const float b0 = nodal[n_nodes + i0];
            const float b1 = nodal[n_nodes + i1];
            const float b2 = nodal[n_nodes + i2];
            u0[i] = fmaf(s[i][0], a0, fmaf(s[i][1], a1, s[i][2] * a2));
            u1[i] = fmaf(s[i][0], b0, fmaf(s[i][1], b1, s[i][2] * b2));
        }

        const v4f o0 = { u0[0], u0[1], u0[2], u0[3] };
        const v4f o1 = { u1[0], u1[1], u1[2], u1[3] };
        // Write-once output: non-temporal b128 stores, keep L2 for the tables.
        __builtin_nontemporal_store(o0, (v4f*)(out + p));
        __builtin_nontemporal_store(o1, (v4f*)(out + n_pts + p));
    }

    // Scalar tail (n_pts not divisible by 4); no-op for the reference sizes.
    for (; p < n_pts; ++p) {
        const int ccc = cid[p];
        const int* cr = conn + (size_t)ccc * 3;
        const int i0 = cr[0] - 1, i1 = cr[1] - 1, i2 = cr[2] - 1;
        const float s0 = shape[(size_t)p * 3 + 0];
        const float s1 = shape[(size_t)p * 3 + 1];
        const float s2 = shape[(size_t)p * 3 + 2];
        out[p]         = fmaf(s0, nodal[i0], fmaf(s1, nodal[i1], s2 * nodal[i2]));
        out[n_pts + p] = fmaf(s0, nodal[n_nodes + i0],
                         fmaf(s1, nodal[n_nodes + i1], s2 * nodal[n_nodes + i2]));
    }
}

extern "C" void kernel_launch(void* const* d_in, const int* in_sizes, int n_in,
                              void* d_out, int out_size, void* d_ws, size_t ws_size,
                              hipStream_t stream) {
    const float* nodal = (const float*)d_in[0];   // 2 * n_nodes
    const float* shape = (const float*)d_in[1];   // n_pts * 3
    const int*   conn  = (const int*)d_in[2];     // n_elem * 3
    const int*   cid   = (const int*)d_in[3];     // n_pts
    float*       out   = (float*)d_out;           // 2 * n_pts

    const int n_pts   = in_sizes[3];
    const int n_nodes = in_sizes[0] / 2;

    const int block = 256;                        // 8 wave32 waves
    const int pts_per_block = block * 4;
    int blocks = (n_pts + pts_per_block - 1) / pts_per_block;
    if (blocks > 2048) blocks = 2048;             // persistent grid-stride
    if (blocks < 1) blocks = 1;

    interp2d_lin_kernel<<<blocks, block, 0, stream>>>(
        nodal, shape, conn, cid, out, n_pts, n_nodes);
}